// ImprovedGCN_3367254360510
// MI455X (gfx1250) — compile-verified
//
#include <hip/hip_runtime.h>
#include <hip/hip_bf16.h>
#include <cstddef>

#define IN_C   128
#define HID    32
#define HID2   64
#define OUT_C  64
#define BN_EPS 1e-5f

typedef float v2f __attribute__((ext_vector_type(2)));
typedef float v8f __attribute__((ext_vector_type(8)));

#ifndef __has_builtin
#define __has_builtin(x) 0
#endif

// ---------------------------------------------------------------------------
// WMMA GEMM + fused GCN self-loop epilogue:
//   Y[m][c]   = (A @ W)[m][c]                       (needed as gather source)
//   Agg[m][c] = Y[m][c] * dinv[m]^2 + bias[c]       (self-loop + bias init,
//                                                    scatter target)
// W (<=16KB) staged into LDS via CDNA5 async global->LDS (ASYNCcnt); B
// fragments then come from LDS. One wave32 owns a 16-row strip and iterates
// all TN = Cout/16 column tiles, reusing each A fragment TN times.
//
// ISA 7.12.2 fp32 fragment maps (g = lane>>4, r = lane&15):
//   A 16x4 : row M=r, VGPR0 K=2g, VGPR1 K=2g+1
//   B 4x16 : col N=r, VGPR0 K=2g, VGPR1 K=2g+1
//   C/D    : VGPR j -> row M=j+8g, col N=r
// ---------------------------------------------------------------------------
template <int K, int Cout>
__global__ __launch_bounds__(256) void gemm_wmma(const float* __restrict__ A,
                                                 const float* __restrict__ W,
                                                 const float* __restrict__ dinv,
                                                 const float* __restrict__ bias,
                                                 float* __restrict__ Y,
                                                 float* __restrict__ Agg, int N) {
  constexpr int TN = Cout / 16;
  __shared__ __align__(16) float sW[K * Cout];

  // Cooperative async copy of W into LDS: 16 bytes per lane-op.
  for (int i = threadIdx.x * 4; i < K * Cout; i += blockDim.x * 4) {
#if defined(__gfx1250__)
    unsigned lds_addr = (unsigned)(size_t)(&sW[i]);
    unsigned long long gaddr = (unsigned long long)(size_t)(W + i);
    asm volatile("global_load_async_to_lds_b128 %0, %1, off"
                 :: "v"(lds_addr), "v"(gaddr) : "memory");
#else
    ((float4*)(&sW[i]))[0] = ((const float4*)(W + i))[0];
#endif
  }
#if defined(__gfx1250__)
#if __has_builtin(__builtin_amdgcn_s_wait_asynccnt)
  __builtin_amdgcn_s_wait_asynccnt(0);
#else
  asm volatile("s_wait_asynccnt 0" ::: "memory");
#endif
#endif
  __syncthreads();

  const int lane = threadIdx.x & 31;
  const int wave = threadIdx.x >> 5;
  const int tilesM = (N + 15) >> 4;
  const int tm = blockIdx.x * (blockDim.x >> 5) + wave;
  if (tm >= tilesM) return;                 // wave-uniform exit, after barrier

  const int g = lane >> 4;
  const int r = lane & 15;
  int arow = tm * 16 + r;
  if (arow >= N) arow = N - 1;              // clamp; clamped rows never stored
  const float* __restrict__ Ar = A + (size_t)arow * K;
  const int kb = 2 * g;                     // per-half-wave K offset (0 or 2)

  v8f acc[TN];
#pragma unroll
  for (int t = 0; t < TN; ++t)
    acc[t] = (v8f){0.f, 0.f, 0.f, 0.f, 0.f, 0.f, 0.f, 0.f};

#pragma unroll 4
  for (int k = 0; k < K; k += 4) {
    v2f a;
    a.x = Ar[k + kb];
    a.y = Ar[k + kb + 1];
#pragma unroll
    for (int t = 0; t < TN; ++t) {
      v2f b;
      b.x = sW[(k + kb) * Cout + t * 16 + r];
      b.y = sW[(k + kb + 1) * Cout + t * 16 + r];
      acc[t] = __builtin_amdgcn_wmma_f32_16x16x4_f32(false, a, false, b,
                                                     (short)0, acc[t], false, false);
    }
  }

  // Epilogue: per-lane row set is m(j) = tm*16 + j + 8g, column = t*16 + r.
  float bb[TN];
#pragma unroll
  for (int t = 0; t < TN; ++t) bb[t] = bias[t * 16 + r];

  float dv2[8];
#pragma unroll
  for (int j = 0; j < 8; ++j) {
    int m = tm * 16 + j + g * 8;
    int mc = (m < N) ? m : (N - 1);
    float d = dinv[mc];
    dv2[j] = d * d;
  }

  const size_t rowBase = (size_t)(tm * 16 + g * 8) * Cout + r;
  if (tm * 16 + 16 <= N) {                  // full tile: unguarded stores
#pragma unroll
    for (int t = 0; t < TN; ++t) {
      size_t p = rowBase + t * 16;
#pragma unroll
      for (int j = 0; j < 8; ++j, p += Cout) {
        float h = acc[t][j];
        Y[p]   = h;
        Agg[p] = h * dv2[j] + bb[t];
      }
    }
  } else {                                  // ragged tail tile
#pragma unroll
    for (int t = 0; t < TN; ++t) {
      size_t p = rowBase + t * 16;
#pragma unroll
      for (int j = 0; j < 8; ++j, p += Cout) {
        int m = tm * 16 + j + g * 8;
        if (m < N) {
          float h = acc[t][j];
          Y[p]   = h;
          Agg[p] = h * dv2[j] + bb[t];
        }
      }
    }
  }
}

// ---------------------------------------------------------------------------
// Graph normalization (computed once, reused by all three layers)
// ---------------------------------------------------------------------------
__global__ void deg_init(float* __restrict__ deg, int N) {
  int i = blockIdx.x * blockDim.x + threadIdx.x;
  if (i < N) deg[i] = 1.0f;                 // self loop
}

__global__ void deg_accum(const int* __restrict__ dst, float* __restrict__ deg, int E) {
  int e = blockIdx.x * blockDim.x + threadIdx.x;
  if (e < E) atomicAdd(&deg[dst[e]], 1.0f);
}

__global__ void deg_rsqrt(float* __restrict__ deg, int N) {
  int i = blockIdx.x * blockDim.x + threadIdx.x;
  if (i < N) deg[i] = rsqrtf(fmaxf(deg[i], 1e-12f));
}

// Per-edge symmetric norm, computed once: norm[e] = dinv[src]*dinv[dst]
__global__ void edge_norm(const int* __restrict__ src, const int* __restrict__ dst,
                          const float* __restrict__ dinv, float* __restrict__ norm, int E) {
  int e = blockIdx.x * blockDim.x + threadIdx.x;
  if (e < E) norm[e] = dinv[src[e]] * dinv[dst[e]];
}

// Per-(edge, 4-channel-group): gather float4 of h[src], scale by precomputed
// edge norm, atomic scatter-add into agg[dst]. C is compile-time -> div/mod
// become shift/mask.
template <int C>
__global__ __launch_bounds__(256) void edge_scatter(const int* __restrict__ src,
                                                    const int* __restrict__ dst,
                                                    const float* __restrict__ norm,
                                                    const float* __restrict__ h,
                                                    float* __restrict__ agg, int E) {
  constexpr int gpe = C >> 2;               // float4 groups per edge (power of 2)
  long long idx = (long long)blockIdx.x * blockDim.x + threadIdx.x;
  if (idx >= (long long)E * gpe) return;
  int e = (int)(idx / gpe);                 // constant divisor -> shift
  int q = (int)(idx % gpe);
  int s = src[e];
  int d = dst[e];
  float nrm = norm[e];
  const float4 v = *(const float4*)(h + (size_t)s * C + q * 4);
  float* o = agg + (size_t)d * C + q * 4;
  atomicAdd(o + 0, v.x * nrm);
  atomicAdd(o + 1, v.y * nrm);
  atomicAdd(o + 2, v.z * nrm);
  atomicAdd(o + 3, v.w * nrm);
}

// out = relu( (agg - mean) * rsqrt(var + eps) * gamma + beta )
template <int C>
__global__ __launch_bounds__(256) void bn_relu(const float* __restrict__ agg,
                                               const float* __restrict__ gam,
                                               const float* __restrict__ bet,
                                               const float* __restrict__ mu,
                                               const float* __restrict__ var,
                                               float* __restrict__ out, int N) {
  long long idx = (long long)blockIdx.x * blockDim.x + threadIdx.x;
  if (idx >= (long long)N * C) return;
  int c = (int)(idx & (C - 1));             // constant mask
  float y = (agg[idx] - mu[c]) * rsqrtf(var[c] + BN_EPS) * gam[c] + bet[c];
  out[idx] = fmaxf(y, 0.0f);
}

// ---------------------------------------------------------------------------
extern "C" void kernel_launch(void* const* d_in, const int* in_sizes, int n_in,
                              void* d_out, int out_size, void* d_ws, size_t ws_size,
                              hipStream_t stream) {
  const float* x   = (const float*)d_in[0];
  const int*   ei  = (const int*)d_in[1];
  const float* W1  = (const float*)d_in[2];
  const float* b1  = (const float*)d_in[3];
  const float* W2  = (const float*)d_in[4];
  const float* b2  = (const float*)d_in[5];
  const float* W3  = (const float*)d_in[6];
  const float* b3  = (const float*)d_in[7];
  const float* g1  = (const float*)d_in[8];
  const float* be1 = (const float*)d_in[9];
  const float* m1  = (const float*)d_in[10];
  const float* v1  = (const float*)d_in[11];
  const float* g2  = (const float*)d_in[12];
  const float* be2 = (const float*)d_in[13];
  const float* m2  = (const float*)d_in[14];
  const float* v2  = (const float*)d_in[15];

  const int N = in_sizes[0] / IN_C;
  const int E = in_sizes[1] / 2;
  const int* src = ei;
  const int* dst = ei + E;

  // Workspace (floats): dinv[N] | norm[E] | buf1[N*64] | buf2[N*64] | buf3[N*64]
  float* dinv = (float*)d_ws;
  float* norm = dinv + N;
  float* buf1 = norm + E;
  float* buf2 = buf1 + (size_t)N * 64;
  float* buf3 = buf2 + (size_t)N * 64;
  float* out  = (float*)d_out;

  const int B = 256;
  auto cdiv = [](long long a, long long b) { return (int)((a + b - 1) / b); };

  // --- normalization (once) ---
  deg_init <<<cdiv(N, B), B, 0, stream>>>(dinv, N);
  deg_accum<<<cdiv(E, B), B, 0, stream>>>(dst, dinv, E);
  deg_rsqrt<<<cdiv(N, B), B, 0, stream>>>(dinv, N);
  edge_norm<<<cdiv(E, B), B, 0, stream>>>(src, dst, dinv, norm, E);

  const int tilesM = (N + 15) / 16;
  const int gemmBlocks = cdiv(tilesM, 8);   // 8 waves/block, one 16-row strip each

  // --- layer 1: h1 = x@W1 -> buf1; agg init fused -> buf2; scatter; bn+relu -> buf1
  gemm_wmma<IN_C, HID><<<gemmBlocks, B, 0, stream>>>(x, W1, dinv, b1, buf1, buf2, N);
  edge_scatter<HID><<<cdiv((long long)E * (HID / 4), B), B, 0, stream>>>(src, dst, norm, buf1, buf2, E);
  bn_relu<HID><<<cdiv((long long)N * HID, B), B, 0, stream>>>(buf2, g1, be1, m1, v1, buf1, N);

  // --- layer 2: h2 = h1@W2 -> buf3; agg init fused -> buf2; scatter; bn+relu -> buf1
  gemm_wmma<HID, HID2><<<gemmBlocks, B, 0, stream>>>(buf1, W2, dinv, b2, buf3, buf2, N);
  edge_scatter<HID2><<<cdiv((long long)E * (HID2 / 4), B), B, 0, stream>>>(src, dst, norm, buf3, buf2, E);
  bn_relu<HID2><<<cdiv((long long)N * HID2, B), B, 0, stream>>>(buf2, g2, be2, m2, v2, buf1, N);

  // --- layer 3: h3 = h1'@W3 -> buf3; agg init fused straight into d_out; scatter into d_out
  gemm_wmma<HID2, OUT_C><<<gemmBlocks, B, 0, stream>>>(buf1, W3, dinv, b3, buf3, out, N);
  edge_scatter<OUT_C><<<cdiv((long long)E * (OUT_C / 4), B), B, 0, stream>>>(src, dst, norm, buf3, out, E);
}